// SemanticBaseline_8306466751172
// MI455X (gfx1250) — compile-verified
//
#include <hip/hip_runtime.h>
#include <cstdint>
#include <math.h>

#define BB   2048
#define KK   24
#define AA   2000
#define A4   (AA / 4)      // 500 float4 per row
#define LAMV 0.5f
#define EPSV 1e-8f

typedef float v4f __attribute__((ext_vector_type(4)));

__device__ __forceinline__ v4f load_stream_v4(const v4f* p) {
#if __has_builtin(__builtin_nontemporal_load)
    return __builtin_nontemporal_load(p);   // th:TH_LOAD_NT -> don't evict emb from L2
#else
    return *p;
#endif
}

// ---------------------------------------------------------------------------
// One block per batch element b.  256 threads = 8 wave32s.
// Phase 1: async-stage sim row emb[aid[b], :] (8 KB) into LDS (ASYNCcnt path).
// Phase 2: each wave processes rows k = wave, wave+8, wave+16.
//          Row slice (64 elems/lane) is pulled into registers in one
//          non-temporal b128 load clause, max is a branch-free tree, then 64
//          independent v_exp_f32 feed FMA trees for (sum, weighted-sum).
//          a_knns is read from HBM exactly once, bypassing L2 retention.
// Phase 3: wave 0 does the K=24 softmax and writes the output.
// ---------------------------------------------------------------------------
__global__ __launch_bounds__(256)
void knn_score_kernel(const float* __restrict__ a_knns,
                      const float* __restrict__ emb,
                      const int*   __restrict__ aids,
                      float*       __restrict__ out)
{
    __shared__ __align__(16) float sim[AA];
    __shared__ float sscores[KK];
    __shared__ float sdiag;

    const int b    = blockIdx.x;
    const int tid  = threadIdx.x;
    const int lane = tid & 31;
    const int wave = tid >> 5;

    const int aid = aids[b];
    const float* simg = emb + (size_t)aid * AA;

    // ---- Phase 1: async global -> LDS staging of the sim row ----
    {
        const uint32_t lds0 = (uint32_t)(uintptr_t)(&sim[0]);
        int i4 = tid;                                  // 0..255
        {
            uint64_t ga = (uint64_t)(uintptr_t)(simg + i4 * 4);
            uint32_t la = lds0 + (uint32_t)(i4 * 16);
            asm volatile("global_load_async_to_lds_b128 %0, %1, off"
                         :: "v"(la), "v"(ga) : "memory");
        }
        i4 = tid + 256;                                // 256..499
        if (i4 < A4) {
            uint64_t ga = (uint64_t)(uintptr_t)(simg + i4 * 4);
            uint32_t la = lds0 + (uint32_t)(i4 * 16);
            asm volatile("global_load_async_to_lds_b128 %0, %1, off"
                         :: "v"(la), "v"(ga) : "memory");
        }
        if (tid == 0) sdiag = simg[aid];               // emb[aid, aid]
#if __has_builtin(__builtin_amdgcn_s_wait_asynccnt)
        __builtin_amdgcn_s_wait_asynccnt(0);
#else
        asm volatile("s_wait_asynccnt 0x0" ::: "memory");
#endif
    }
    __syncthreads();

    const float diagv = sdiag;
    const v4f* sim4 = reinterpret_cast<const v4f*>(sim);

    // ---- Phase 2: two-pass (regs) softmax + weighted sum, 3 rows per wave ----
    for (int k = wave; k < KK; k += 8) {
        const float* row  = a_knns + ((size_t)b * KK + k) * AA;
        const v4f*   row4 = reinterpret_cast<const v4f*>(row);

        // Load pass: 16 non-temporal b128 loads as one clause; -INF pad.
        v4f xv[16];
        #pragma unroll
        for (int it = 0; it < 16; ++it) {
            const int i4 = it * 32 + lane;
            if (i4 < A4) xv[it] = load_stream_v4(&row4[i4]);
            else         xv[it] = (v4f)(-INFINITY);
        }

        // Branch-free max tree (registers), then wave32 butterfly.
        float m = -INFINITY;
        #pragma unroll
        for (int it = 0; it < 16; ++it)
            m = fmaxf(m, fmaxf(fmaxf(xv[it].x, xv[it].y),
                               fmaxf(xv[it].z, xv[it].w)));
        #pragma unroll
        for (int off = 16; off > 0; off >>= 1)
            m = fmaxf(m, __shfl_xor(m, off, 32));

        // Exp pass: 64 independent v_exp_f32 per lane, FMA accumulation.
        // Padded slots: exp(-INF - m) == 0 kills both s and w contributions.
        float s = 0.f, w = 0.f;
        #pragma unroll
        for (int it = 0; it < 16; ++it) {
            const int i4 = it * 32 + lane;
            const v4f sv = sim4[(i4 < A4) ? i4 : 0];
            const float e0 = __expf(xv[it].x - m);
            const float e1 = __expf(xv[it].y - m);
            const float e2 = __expf(xv[it].z - m);
            const float e3 = __expf(xv[it].w - m);
            s += (e0 + e1) + (e2 + e3);
            w += e0 * sv.x + e1 * sv.y + e2 * sv.z + e3 * sv.w;
        }

        // Plain butterfly sums (m is already global to the wave).
        #pragma unroll
        for (int off = 16; off > 0; off >>= 1) {
            s += __shfl_xor(s, off, 32);
            w += __shfl_xor(w, off, 32);
        }

        if (lane == 0) {
            const float x_aid  = row[aid];
            const float nb_aid = __expf(x_aid - m) / s;
            const float wsv    = w / s - nb_aid * diagv;
            sscores[k] = LAMV * wsv - (1.0f - LAMV) * __logf(nb_aid + EPSV);
        }
    }
    __syncthreads();

    // ---- Phase 3: softmax over K = 24, done by wave 0 ----
    if (tid < 32) {
        const float sc = (lane < KK) ? sscores[lane] : -INFINITY;
        float m = sc;
        #pragma unroll
        for (int off = 16; off > 0; off >>= 1)
            m = fmaxf(m, __shfl_xor(m, off, 32));
        const float e = (lane < KK) ? __expf(sc - m) : 0.f;
        float ssum = e;
        #pragma unroll
        for (int off = 16; off > 0; off >>= 1)
            ssum += __shfl_xor(ssum, off, 32);
        if (lane < KK) out[(size_t)b * KK + lane] = e / ssum;
    }
}

extern "C" void kernel_launch(void* const* d_in, const int* in_sizes, int n_in,
                              void* d_out, int out_size, void* d_ws, size_t ws_size,
                              hipStream_t stream)
{
    (void)in_sizes; (void)n_in; (void)out_size; (void)d_ws; (void)ws_size;
    const float* a_knns = (const float*)d_in[0];
    const float* emb    = (const float*)d_in[1];
    const int*   aids   = (const int*)d_in[2];
    float*       out    = (float*)d_out;

    hipLaunchKernelGGL(knn_score_kernel, dim3(BB), dim3(256), 0, stream,
                       a_knns, emb, aids, out);
}